// GraphIsomorphismNetwork_43688407335392
// MI455X (gfx1250) — compile-verified
//
#include <hip/hip_runtime.h>
#include <hip/hip_bf16.h>

typedef __attribute__((ext_vector_type(2))) float v2f;
typedef __attribute__((ext_vector_type(8))) float v8f;

#define M_NODES 100000
#define N_EDGES_C 800000

// ---------------------------------------------------------------------------
// GEMM: C[M,N] = A[M,K] @ B[K,N] + bias, optional ReLU.  fp32 WMMA 16x16x4.
// N, K are compile-time so the hot loop has zero guards and immediate offsets.
// One wave per 16-row tile (M = 6250 * 16 exactly), 16x64 strip per n-step.
// A frag (16x4):  lanes 0-15 -> (K=k, K=k+1), lanes 16-31 -> (K=k+2, K=k+3)
// B frag (4x16):  lanes 0-15 -> rows k,k+1 ; lanes 16-31 -> rows k+2,k+3
// C/D (16x16 f32): vgpr i, lanes 0-15 -> row i col=lane; lanes 16-31 -> row i+8
// ---------------------------------------------------------------------------
template<int N, int K>
__global__ __launch_bounds__(256)
void gemm_bias_kernel(const float* __restrict__ A, const float* __restrict__ B,
                      const float* __restrict__ bias, float* __restrict__ C,
                      int relu)
{
    constexpr int KMAIN = K & ~3;            // 84 for K=86; K when K%4==0

    const int lane = threadIdx.x & 31;
    const int wave = threadIdx.x >> 5;
    const int half = lane >> 4;   // 0: lanes 0-15, 1: lanes 16-31
    const int l15  = lane & 15;
    const int m_tile = blockIdx.x * 8 + wave;
    if (m_tile * 16 >= M_NODES) return;      // wave-uniform exit (EXEC stays all-1)
    const int m_base = m_tile * 16;
    const float* __restrict__ Arow = A + (size_t)(m_base + l15) * K;

    for (int n0 = 0; n0 < N; n0 += 64) {
        v8f acc0 = {}, acc1 = {}, acc2 = {}, acc3 = {};

        // per-lane streaming pointers; strides are compile-time constants
        const float* __restrict__ ap = Arow + 2 * half;                      // +4 per step
        const float* __restrict__ bp = B + (size_t)(2 * half) * N + n0 + l15; // +4*N per step

        #pragma unroll 2
        for (int k = 0; k < KMAIN; k += 4) {
            v2f a = *(const v2f*)ap;         // 8B-aligned for K in {86,128,256}
            v2f b0, b1, b2, b3;
            b0.x = bp[0];      b0.y = bp[N];
            b1.x = bp[16];     b1.y = bp[N + 16];
            b2.x = bp[32];     b2.y = bp[N + 32];
            b3.x = bp[48];     b3.y = bp[N + 48];
            acc0 = __builtin_amdgcn_wmma_f32_16x16x4_f32(false, a, false, b0, (short)0, acc0, false, false);
            acc1 = __builtin_amdgcn_wmma_f32_16x16x4_f32(false, a, false, b1, (short)0, acc1, false, false);
            acc2 = __builtin_amdgcn_wmma_f32_16x16x4_f32(false, a, false, b2, (short)0, acc2, false, false);
            acc3 = __builtin_amdgcn_wmma_f32_16x16x4_f32(false, a, false, b3, (short)0, acc3, false, false);
            ap += 4;
            bp += 4 * N;
        }

        if constexpr ((K & 3) != 0) {
            // Remainder K-step: clamp addresses (unconditional loads, no EXEC
            // divergence) and zero invalid elements with data-side selects.
            const int  ka  = KMAIN + 2 * half;
            const bool ok0 = (ka     < K);
            const bool ok1 = (ka + 1 < K);
            const int  ia0 = ok0 ? ka       : (K - 1);
            const int  ia1 = ok1 ? (ka + 1) : (K - 1);
            v2f a;
            a.x = ok0 ? Arow[ia0] : 0.0f;
            a.y = ok1 ? Arow[ia1] : 0.0f;
            const float* __restrict__ Br0 = B + (size_t)ia0 * N + n0 + l15;
            const float* __restrict__ Br1 = B + (size_t)ia1 * N + n0 + l15;
            v2f b0, b1, b2, b3;
            b0.x = ok0 ? Br0[0]  : 0.0f;  b0.y = ok1 ? Br1[0]  : 0.0f;
            b1.x = ok0 ? Br0[16] : 0.0f;  b1.y = ok1 ? Br1[16] : 0.0f;
            b2.x = ok0 ? Br0[32] : 0.0f;  b2.y = ok1 ? Br1[32] : 0.0f;
            b3.x = ok0 ? Br0[48] : 0.0f;  b3.y = ok1 ? Br1[48] : 0.0f;
            acc0 = __builtin_amdgcn_wmma_f32_16x16x4_f32(false, a, false, b0, (short)0, acc0, false, false);
            acc1 = __builtin_amdgcn_wmma_f32_16x16x4_f32(false, a, false, b1, (short)0, acc1, false, false);
            acc2 = __builtin_amdgcn_wmma_f32_16x16x4_f32(false, a, false, b2, (short)0, acc2, false, false);
            acc3 = __builtin_amdgcn_wmma_f32_16x16x4_f32(false, a, false, b3, (short)0, acc3, false, false);
        }

        const int row_b = m_base + 8 * half;
        const int col0  = n0 + l15;
        const float bb0 = bias[col0];
        const float bb1 = bias[col0 + 16];
        const float bb2 = bias[col0 + 32];
        const float bb3 = bias[col0 + 48];
        #pragma unroll
        for (int i = 0; i < 8; ++i) {
            const size_t rbase = (size_t)(row_b + i) * N;
            float o0 = acc0[i] + bb0;
            float o1 = acc1[i] + bb1;
            float o2 = acc2[i] + bb2;
            float o3 = acc3[i] + bb3;
            if (relu) {
                o0 = fmaxf(o0, 0.0f); o1 = fmaxf(o1, 0.0f);
                o2 = fmaxf(o2, 0.0f); o3 = fmaxf(o3, 0.0f);
            }
            C[rbase + col0]      = o0;
            C[rbase + col0 + 16] = o1;
            C[rbase + col0 + 32] = o2;
            C[rbase + col0 + 48] = o3;
        }
    }
}

// h[i] = (1 + eps) * x[i]
__global__ void scale_init_kernel(const float* __restrict__ x,
                                  const float* __restrict__ eps,
                                  float* __restrict__ h, int n)
{
    int i = blockIdx.x * blockDim.x + threadIdx.x;
    if (i < n) h[i] = (1.0f + eps[0]) * x[i];
}

// h[dst[e]][f] += x[src[e]][f]   (segment_sum over edges)
__global__ __launch_bounds__(256)
void scatter_add_kernel(const float* __restrict__ x, const int* __restrict__ src,
                        const int* __restrict__ dst, float* __restrict__ h, int dim)
{
    long long idx   = (long long)blockIdx.x * blockDim.x + threadIdx.x;
    long long total = (long long)N_EDGES_C * dim;
    if (idx >= total) return;
    int e = (int)(idx / dim);
    int f = (int)(idx % dim);
    int s = src[e], d = dst[e];
    float val = x[(size_t)s * dim + f];
    __hip_atomic_fetch_add(&h[(size_t)d * dim + f], val,
                           __ATOMIC_RELAXED, __HIP_MEMORY_SCOPE_AGENT);
}

__global__ void zero_kernel(float* __restrict__ p, int n)
{
    int i = blockIdx.x * blockDim.x + threadIdx.x;
    if (i < n) p[i] = 0.0f;
}

// per-column sum / sum-of-squares into stats[0..dim) and stats[dim..2*dim)
__global__ __launch_bounds__(256)
void bn_stats_kernel(const float* __restrict__ y, float* __restrict__ stats,
                     int dim, int rows_per_block)
{
    int r0 = blockIdx.x * rows_per_block;
    int r1 = r0 + rows_per_block; if (r1 > M_NODES) r1 = M_NODES;
    if (r0 >= r1) return;
    for (int c = threadIdx.x; c < dim; c += blockDim.x) {
        float s = 0.0f, s2 = 0.0f;
        for (int r = r0; r < r1; ++r) {
            float t = y[(size_t)r * dim + c];
            s += t; s2 += t * t;
        }
        __hip_atomic_fetch_add(&stats[c], s, __ATOMIC_RELAXED, __HIP_MEMORY_SCOPE_AGENT);
        __hip_atomic_fetch_add(&stats[dim + c], s2, __ATOMIC_RELAXED, __HIP_MEMORY_SCOPE_AGENT);
    }
}

// out = relu((y - mean) * rsqrt(var + eps) * g + bt) + res
__global__ __launch_bounds__(256)
void bn_apply_kernel(const float* __restrict__ y, const float* __restrict__ stats,
                     const float* __restrict__ g, const float* __restrict__ bt,
                     const float* __restrict__ res, float* __restrict__ out,
                     int dim, int n)
{
    int i = blockIdx.x * blockDim.x + threadIdx.x;
    if (i >= n) return;
    int c = i % dim;
    const float inv_m = 1.0f / (float)M_NODES;
    float m   = stats[c] * inv_m;
    float var = stats[dim + c] * inv_m - m * m;
    float r   = rsqrtf(var + 1e-5f);
    float v   = (y[i] - m) * r * g[c] + bt[c];
    v = fmaxf(v, 0.0f);
    out[i] = v + res[i];
}

static inline void launch_gemm(const float* A, const float* B, const float* bias,
                               float* C, int N, int K, int relu, hipStream_t stream)
{
    const int grid = (6250 + 7) / 8;  // 782 blocks of 8 waves (16 rows each)
    if (N == 128 && K == 86) {
        gemm_bias_kernel<128, 86><<<grid, 256, 0, stream>>>(A, B, bias, C, relu);
    } else if (N == 256 && K == 128) {
        gemm_bias_kernel<256, 128><<<grid, 256, 0, stream>>>(A, B, bias, C, relu);
    } else if (N == 512 && K == 256) {
        gemm_bias_kernel<512, 256><<<grid, 256, 0, stream>>>(A, B, bias, C, relu);
    }
}

extern "C" void kernel_launch(void* const* d_in, const int* in_sizes, int n_in,
                              void* d_out, int out_size, void* d_ws, size_t ws_size,
                              hipStream_t stream)
{
    (void)in_sizes; (void)n_in; (void)out_size; (void)ws_size;

    const float* v     = (const float*)d_in[0];
    const int*   edges = (const int*)d_in[1];
    const float* We1 = (const float*)d_in[2];
    const float* be1 = (const float*)d_in[3];
    const float* We2 = (const float*)d_in[4];
    const float* be2 = (const float*)d_in[5];
    const float* We3 = (const float*)d_in[6];
    const float* be3 = (const float*)d_in[7];
    const float* eps1 = (const float*)d_in[8];
    const float* W1  = (const float*)d_in[9];
    const float* b1  = (const float*)d_in[10];
    const float* g1  = (const float*)d_in[11];
    const float* bt1 = (const float*)d_in[12];
    const float* eps2 = (const float*)d_in[13];
    const float* W2  = (const float*)d_in[14];
    const float* b2  = (const float*)d_in[15];
    const float* g2  = (const float*)d_in[16];
    const float* bt2 = (const float*)d_in[17];
    const float* eps3 = (const float*)d_in[18];
    const float* W3  = (const float*)d_in[19];
    const float* b3  = (const float*)d_in[20];
    const float* g3  = (const float*)d_in[21];
    const float* bt3 = (const float*)d_in[22];

    const int* src = edges;
    const int* dst = edges + N_EDGES_C;

    // Workspace layout (floats): V1 | V2 | V3 | H | Y | stats
    float* ws = (float*)d_ws;
    const size_t M = M_NODES;
    float* V1 = ws;                 // 100k x 128
    float* V2 = V1 + M * 128;       // 100k x 256
    float* V3 = V2 + M * 256;       // 100k x 512
    float* H  = V3 + M * 512;       // up to 100k x 256 (layer inputs)
    float* Y  = H  + M * 256;       // up to 100k x 512 (pre-BN linear out)
    float* ST = Y  + M * 512;       // 2 * 512 stats

    auto gin_layer = [&](const float* x, int din, const float* epsp,
                         const float* W, const float* b,
                         const float* g, const float* bt,
                         const float* res, float* out, int dout) {
        const int n_in = M_NODES * din;
        scale_init_kernel<<<(n_in + 255) / 256, 256, 0, stream>>>(x, epsp, H, n_in);
        const long long te = (long long)N_EDGES_C * din;
        scatter_add_kernel<<<(int)((te + 255) / 256), 256, 0, stream>>>(x, src, dst, H, din);
        launch_gemm(H, W, b, Y, dout, din, 0, stream);
        zero_kernel<<<4, 256, 0, stream>>>(ST, 2 * dout);
        const int rpb = (M_NODES + 239) / 240;
        bn_stats_kernel<<<240, 256, 0, stream>>>(Y, ST, dout, rpb);
        const int n_out = M_NODES * dout;
        bn_apply_kernel<<<(n_out + 255) / 256, 256, 0, stream>>>(Y, ST, g, bt, res, out, dout, n_out);
    };

    // Encoder chain (must complete before xv overwrites V1/V2)
    launch_gemm(v,  We1, be1, V1, 128, 86,  1, stream);  // v1
    launch_gemm(V1, We2, be2, V2, 256, 128, 1, stream);  // v2
    launch_gemm(V2, We3, be3, V3, 512, 256, 1, stream);  // v3

    // GIN stack with fused residuals; xv1/xv2 written in place over V1/V2
    gin_layer(v,  86,  eps1, W1, b1, g1, bt1, V1, V1, 128);            // xv1 = x1 + v1
    gin_layer(V1, 128, eps2, W2, b2, g2, bt2, V2, V2, 256);            // xv2 = x2 + v2
    gin_layer(V2, 256, eps3, W3, b3, g3, bt3, V3, (float*)d_out, 512); // out = x3 + v3
}